// SupConLoss_66838281061046
// MI455X (gfx1250) — compile-verified
//
#include <hip/hip_runtime.h>
#include <hip/hip_bf16.h>
#include <math.h>

// ---------------------------------------------------------------------------
// SupConLoss for MI455X (gfx1250, wave32, WMMA)
//   inputs : [8192,128] f32, targets : [8192] int
//   out    : scalar f32 = 0.1*CE + 0.9*SupCon
// ---------------------------------------------------------------------------

typedef float v2f __attribute__((ext_vector_type(2)));
typedef float v4f __attribute__((ext_vector_type(4)));
typedef float v8f __attribute__((ext_vector_type(8)));

#define B_ROWS   8192
#define DIMS     128
#define NLAB     128
#define INV_TEMP (1.0f / 0.3f)
#define LAMW     0.9f
#define COL_SPLITS 4
#define TILES_PER_SPLIT 128   // 512 col tiles / 4 splits
#define LDS_STRIDE 132        // 132 % 64 == 4 -> conflict-free b64 frag loads

#if __has_builtin(__builtin_amdgcn_sched_barrier)
#define SCHED_FENCE() __builtin_amdgcn_sched_barrier(0)
#else
#define SCHED_FENCE()
#endif

// one-exp online logsumexp merge of a single new value `v` into (m, s)
__device__ __forceinline__ void online_update(float& m, float& s, float v) {
  const float d = v - m;
  const float e = __expf(-fabsf(d));   // single v_exp_f32
  const bool up = d > 0.0f;
  s = up ? __fmaf_rn(s, e, 1.0f) : (s + e);
  m = up ? v : m;
}

// one-exp merge of two (m, s) states into (m1, s1)
__device__ __forceinline__ void state_merge(float& m1, float& s1,
                                            float m2, float s2) {
  const float d = m2 - m1;
  const float e = __expf(-fabsf(d));
  const bool up = d > 0.0f;
  s1 = up ? __fmaf_rn(s1, e, s2) : __fmaf_rn(s2, e, s1);
  m1 = up ? m2 : m1;
}

// --------------------------------------------------------------------------
// Kernel 1: per-class feature sums c_l[128][128] and counts[128].
// One block per label, one thread per dim. Fixed summation order (determ.)
// --------------------------------------------------------------------------
__global__ __launch_bounds__(DIMS) void class_sums_kernel(
    const float* __restrict__ X, const int* __restrict__ tg,
    float* __restrict__ csum, float* __restrict__ counts) {
  const int l = blockIdx.x;
  const int d = threadIdx.x;
  float s = 0.0f;
  int cnt = 0;
  for (int r = 0; r < B_ROWS; ++r) {
    if (tg[r] == l) { s += X[r * DIMS + d]; ++cnt; }
  }
  csum[l * DIMS + d] = s;
  if (d == 0) counts[l] = (float)cnt;
}

// --------------------------------------------------------------------------
// Kernel 2: partial row-wise online logsumexp of sim = X X^T / TEMP,
// diagonal excluded. grid = (64 row blocks, 4 column splits).
// Block = 256 threads = 8 waves; each wave owns one 16-row tile and sweeps
// 128 column tiles (2048 columns) with V_WMMA_F32_16X16X4_F32.
// sched_barrier pins the 16 batched ds_load_2addr_b64 fragment loads ahead
// of an unbroken 32-WMMA accumulate chain (one dscnt wait per tile).
// --------------------------------------------------------------------------
__global__ __launch_bounds__(256) void lse_partial_kernel(
    const float* __restrict__ X,
    float* __restrict__ pmax, float* __restrict__ psum) {
  __shared__ float Bs[16 * LDS_STRIDE];

  const int tid  = threadIdx.x;
  const int lane = tid & 31;
  const int wave = tid >> 5;
  const int row_base = blockIdx.x * 128 + wave * 16;
  const int split    = blockIdx.y;

  const int n    = lane & 15;    // column within tile (and row M for A frag)
  const int half = lane >> 4;    // which K pair inside a K=4 chunk

  // A fragments resident: a[k] = X[row_base+n][4k + 2*half + {0,1}]
  v2f a[32];
  {
    const float* arow = X + (size_t)(row_base + n) * DIMS + 2 * half;
#pragma unroll
    for (int k = 0; k < 32; ++k) a[k] = *(const v2f*)(arow + 4 * k);
  }

  float mx[8], ss[8];
#pragma unroll
  for (int r = 0; r < 8; ++r) { mx[r] = -1.0e30f; ss[r] = 0.0f; }

  for (int ct = 0; ct < TILES_PER_SPLIT; ++ct) {
    const int col_base = (split * TILES_PER_SPLIT + ct) * 16;

    __syncthreads();
    // stage 16x128 column panel into LDS (padded stride), 8 dwords/thread
#pragma unroll
    for (int i = 0; i < 8; ++i) {
      int idx = tid + i * 256;          // 0..2047
      int rr = idx >> 7, cc = idx & 127;
      Bs[rr * LDS_STRIDE + cc] = X[(size_t)(col_base + rr) * DIMS + cc];
    }
    __syncthreads();

    // Preload ALL B fragments: 16 back-to-back ds_load_2addr_b64.
    const float* bl = Bs + n * LDS_STRIDE + 2 * half;
    v2f b[32];
#pragma unroll
    for (int k = 0; k < 32; ++k) b[k] = *(const v2f*)(bl + 4 * k);

    // Keep the scheduler from sinking the loads back into the WMMA chain.
    SCHED_FENCE();

    v8f acc = {0.0f, 0.0f, 0.0f, 0.0f, 0.0f, 0.0f, 0.0f, 0.0f};
#pragma unroll
    for (int k = 0; k < 32; ++k) {
      acc = __builtin_amdgcn_wmma_f32_16x16x4_f32(
          /*neg_a=*/false, a[k], /*neg_b=*/false, b[k],
          /*c_mod=*/(short)0, acc, /*reuse_a=*/false, /*reuse_b=*/false);
    }

    SCHED_FENCE();

    const bool diag_tile = (col_base == row_base);
#pragma unroll
    for (int r = 0; r < 8; ++r) {
      // acc[r] is element (M = r + 8*half, N = n) of the 16x16 tile
      const bool isdiag = diag_tile && (n == r + 8 * half);
      const float val = isdiag ? -3.0e38f : acc[r] * INV_TEMP;
      online_update(mx[r], ss[r], val);
    }
  }

  // merge the 16 lanes of each half-wave (one matrix row per (r, half))
#pragma unroll
  for (int r = 0; r < 8; ++r) {
    float M = mx[r], S = ss[r];
#pragma unroll
    for (int off = 1; off < 16; off <<= 1) {
      const float M2 = __shfl_xor(M, off, 32);
      const float S2 = __shfl_xor(S, off, 32);
      state_merge(M, S, M2, S2);
    }
    if (n == 0) {
      const int row = row_base + r + 8 * half;
      pmax[split * B_ROWS + row] = M;
      psum[split * B_ROWS + row] = S;
    }
  }
}

// --------------------------------------------------------------------------
// Kernel 3: per-row finalize. One wave per row.
//   lse_i  = merge of 4 split partials
//   CE_i   = -(x[t] - logsumexp(x))
//   S_i    = (x_i . c_{y_i} - ||x_i||^2) / TEMP   (positive-pair sum trick)
//   scl_i  = npos>0 ? -(S_i - npos*lse_i)/npos : 0
// --------------------------------------------------------------------------
__global__ __launch_bounds__(256) void finalize_rows_kernel(
    const float* __restrict__ X, const int* __restrict__ tg,
    const float* __restrict__ csum, const float* __restrict__ counts,
    const float* __restrict__ pmax, const float* __restrict__ psum,
    float* __restrict__ rowce, float* __restrict__ rowscl) {
  const int lane = threadIdx.x & 31;
  const int wave = threadIdx.x >> 5;
  const int row  = blockIdx.x * 8 + wave;

  const float* x = X + (size_t)row * DIMS;
  const v4f xv = *(const v4f*)(x + lane * 4);

  // ---- CE: logsumexp over this row's 128 logits ----
  float lmax = fmaxf(fmaxf(xv[0], xv[1]), fmaxf(xv[2], xv[3]));
#pragma unroll
  for (int off = 1; off < 32; off <<= 1)
    lmax = fmaxf(lmax, __shfl_xor(lmax, off, 32));
  float lsum = __expf(xv[0] - lmax) + __expf(xv[1] - lmax) +
               __expf(xv[2] - lmax) + __expf(xv[3] - lmax);
#pragma unroll
  for (int off = 1; off < 32; off <<= 1)
    lsum += __shfl_xor(lsum, off, 32);

  const int t = tg[row];

  // ---- dot(x, c_t) and ||x||^2 ----
  const v4f cv = *(const v4f*)(csum + (size_t)t * DIMS + lane * 4);
  float dot  = xv[0] * cv[0] + xv[1] * cv[1] + xv[2] * cv[2] + xv[3] * cv[3];
  float self = xv[0] * xv[0] + xv[1] * xv[1] + xv[2] * xv[2] + xv[3] * xv[3];
#pragma unroll
  for (int off = 1; off < 32; off <<= 1) {
    dot  += __shfl_xor(dot, off, 32);
    self += __shfl_xor(self, off, 32);
  }

  if (lane == 0) {
    // merge lse partials (fixed order)
    float M = pmax[row], S = psum[row];
#pragma unroll
    for (int p = 1; p < COL_SPLITS; ++p)
      state_merge(M, S, pmax[p * B_ROWS + row], psum[p * B_ROWS + row]);
    const float lse_i = M + __logf(S);

    const float ce = -(x[t] - lmax - __logf(lsum));
    const float npos = counts[t] - 1.0f;
    const float Si = (dot - self) * INV_TEMP;
    const float scl = (npos > 0.5f) ? (-(Si - npos * lse_i) / npos) : 0.0f;
    rowce[row]  = ce;
    rowscl[row] = scl;
  }
}

// --------------------------------------------------------------------------
// Kernel 4: deterministic tree reduction -> scalar loss
// --------------------------------------------------------------------------
__global__ __launch_bounds__(256) void reduce_out_kernel(
    const float* __restrict__ rowce, const float* __restrict__ rowscl,
    float* __restrict__ out) {
  __shared__ float s1[256];
  __shared__ float s2[256];
  const int tid = threadIdx.x;
  float a = 0.0f, b = 0.0f;
  for (int i = tid; i < B_ROWS; i += 256) { a += rowce[i]; b += rowscl[i]; }
  s1[tid] = a; s2[tid] = b;
  __syncthreads();
  for (int s = 128; s > 0; s >>= 1) {
    if (tid < s) { s1[tid] += s1[tid + s]; s2[tid] += s2[tid + s]; }
    __syncthreads();
  }
  if (tid == 0)
    out[0] = (1.0f - LAMW) * (s1[0] / (float)B_ROWS) + LAMW * s2[0];
}

// --------------------------------------------------------------------------
// Launch
// --------------------------------------------------------------------------
extern "C" void kernel_launch(void* const* d_in, const int* in_sizes, int n_in,
                              void* d_out, int out_size, void* d_ws, size_t ws_size,
                              hipStream_t stream) {
  (void)in_sizes; (void)n_in; (void)out_size; (void)ws_size;
  const float* X  = (const float*)d_in[0];
  const int*   tg = (const int*)d_in[1];
  float* out = (float*)d_out;

  // workspace layout (floats)
  float* W = (float*)d_ws;
  float* csum   = W;                      // 128*128
  float* counts = csum + NLAB * DIMS;     // 128
  float* pmax   = counts + NLAB;          // 4*8192
  float* psum   = pmax + COL_SPLITS * B_ROWS;
  float* rowce  = psum + COL_SPLITS * B_ROWS;
  float* rowscl = rowce + B_ROWS;

  class_sums_kernel<<<NLAB, DIMS, 0, stream>>>(X, tg, csum, counts);
  lse_partial_kernel<<<dim3(B_ROWS / 128, COL_SPLITS), 256, 0, stream>>>(X, pmax, psum);
  finalize_rows_kernel<<<B_ROWS / 8, 256, 0, stream>>>(X, tg, csum, counts,
                                                       pmax, psum, rowce, rowscl);
  reduce_out_kernel<<<1, 256, 0, stream>>>(rowce, rowscl, out);
}